// Attention_35708358099413
// MI455X (gfx1250) — compile-verified
//
#include <hip/hip_runtime.h>

typedef __attribute__((ext_vector_type(2))) float v2f;
typedef __attribute__((ext_vector_type(8))) float v8f;

#define T_DIM    8192
#define B_DIM    64
#define H_DIM    256
#define N_DIM    128
#define NCHUNK   32
#define TPC      (T_DIM / NCHUNK)     // 256 timesteps per chunk
#define PLANE_F4 ((B_DIM * H_DIM) / 4) // 4096 float4 per (b,d) plane

// ---------------------------------------------------------------------------
// Kernel 1: partial column sums of h over T.  h is [T, B, H] -> plane (B*H)
// contiguous per t.  Each (plane-block, chunk) block streams 1 MB with
// float4 (global_load_b128) loads; 512 blocks saturate HBM.
// ---------------------------------------------------------------------------
__global__ __launch_bounds__(256) void hsum_partial_kernel(
    const float4* __restrict__ h4, float4* __restrict__ part4) {
  const int i4    = blockIdx.x * 256 + threadIdx.x;   // 0..4095 within plane
  const int chunk = blockIdx.y;                       // 0..31
  const float4* p = h4 + (size_t)chunk * TPC * PLANE_F4 + i4;
  float4 acc = {0.f, 0.f, 0.f, 0.f};
  for (int t = 0; t < TPC; ++t) {
    float4 v = p[(size_t)t * PLANE_F4];
    acc.x += v.x; acc.y += v.y; acc.z += v.z; acc.w += v.w;
  }
  part4[chunk * PLANE_F4 + i4] = acc;
}

// ---------------------------------------------------------------------------
// Kernel 2: reduce the 32 partial planes -> hsum[B*H].
// ---------------------------------------------------------------------------
__global__ __launch_bounds__(256) void hsum_reduce_kernel(
    const float4* __restrict__ part4, float4* __restrict__ hsum4) {
  const int i4 = blockIdx.x * 256 + threadIdx.x;      // 0..4095
  float4 acc = {0.f, 0.f, 0.f, 0.f};
  #pragma unroll
  for (int c = 0; c < NCHUNK; ++c) {
    float4 v = part4[c * PLANE_F4 + i4];
    acc.x += v.x; acc.y += v.y; acc.z += v.z; acc.w += v.w;
  }
  hsum4[i4] = acc;
}

// ---------------------------------------------------------------------------
// Kernel 3: sW = s_before @ W^T + bias via V_WMMA_F32_16X16X4_F32 (exact f32).
//   M = B_DIM = 64 rows, N = H_DIM = 256 cols, K = H_DIM = 256.
//   One 16x16 tile per wave; 4x16 = 64 tiles = 64 waves = 8 blocks x 8 waves.
// Fragment layouts per ISA 7.12.2 (32-bit 16x4 A; B mirrored; C/D 8 VGPRs).
// ---------------------------------------------------------------------------
__global__ __launch_bounds__(256) void sw_wmma_kernel(
    const float* __restrict__ S,     // s_before [64,256] row-major
    const float* __restrict__ W,     // W        [256,256] row-major (B[k][n]=W[n][k])
    const float* __restrict__ bias,  // [256]
    float* __restrict__ sW) {        // [64,256]
  const int tid  = threadIdx.x;
  const int lane = tid & 31;
  const int wave = blockIdx.x * 8 + (tid >> 5);  // 0..63
  const int tileM = wave >> 4;                   // 0..3
  const int tileN = wave & 15;                   // 0..15
  const int r    = lane & 15;                    // row/col within half-wave
  const int koff = (lane < 16) ? 0 : 2;          // K offset per A/B layout
  const int mA = tileM * 16 + r;                 // A row this lane feeds
  const int nB = tileN * 16 + r;                 // B col this lane feeds

  v8f acc = {};
  for (int k = 0; k < H_DIM; k += 4) {
    // A frag: VGPR0 = A[m][k+koff], VGPR1 = A[m][k+koff+1]
    v2f a = *(const v2f*)(S + mA * H_DIM + k + koff);
    // B frag: VGPR0 = B[k+koff][n] = W[n][k+koff], VGPR1 = W[n][k+koff+1]
    v2f b = *(const v2f*)(W + nB * H_DIM + k + koff);
    acc = __builtin_amdgcn_wmma_f32_16x16x4_f32(
        /*neg_a=*/false, a, /*neg_b=*/false, b,
        /*c_mod=*/(short)0, acc, /*reuse_a=*/false, /*reuse_b=*/false);
  }

  // D layout: VGPR j -> M = tileM*16 + j (+8 for lanes 16..31), N = r.
  const int nOut  = tileN * 16 + r;
  const int mBase = tileM * 16 + ((lane < 16) ? 0 : 8);
  const float bv  = bias[nOut];
  #pragma unroll
  for (int j = 0; j < 8; ++j) {
    sW[(mBase + j) * H_DIM + nOut] = acc[j] + bv;
  }
}

// ---------------------------------------------------------------------------
// Block-wide sum over 256 threads (8 wave32): shfl_xor intra-wave, LDS across.
// ---------------------------------------------------------------------------
__device__ __forceinline__ float block_reduce_sum(float v, float* red) {
  #pragma unroll
  for (int off = 16; off > 0; off >>= 1) v += __shfl_xor(v, off, 32);
  const int wave = threadIdx.x >> 5;
  const int lane = threadIdx.x & 31;
  __syncthreads();                 // protect red[] from previous round
  if (lane == 0) red[wave] = v;
  __syncthreads();
  return red[0] + red[1] + red[2] + red[3] + red[4] + red[5] + red[6] + red[7];
}

// ---------------------------------------------------------------------------
// Kernel 4: per-batch fused denom + scores + context.
//   denom[b]  = dot(sW[b,:], hsum[b,:])
//   score_n   = dot(sW[b,:], h_sliced[n,b,:]) / denom[b]
//   c_t[b,h]  = sum_n score_n * h_sliced[n,b,h]
// Each h_sliced row is loaded once into registers and reused for both uses.
// ---------------------------------------------------------------------------
__global__ __launch_bounds__(256) void attn_fused_kernel(
    const float* __restrict__ sW,    // [64,256]
    const float* __restrict__ hsum,  // [64,256] (b,d) plane of sum_t h
    const float* __restrict__ hs,    // h_sliced [N, B, H]
    float* __restrict__ out) {       // c_t [B, H]
  __shared__ float red[8];
  const int b   = blockIdx.x;        // 0..63
  const int tid = threadIdx.x;       // 0..255 == hidden index

  const float sw = sW[b * H_DIM + tid];
  const float denom = block_reduce_sum(sw * hsum[b * H_DIM + tid], red);
  const float inv_denom = 1.0f / denom;

  float cacc = 0.f;
  for (int n = 0; n < N_DIM; ++n) {
    const float v = hs[((size_t)n * B_DIM + b) * H_DIM + tid];
    const float s = block_reduce_sum(sw * v, red);
    cacc += (s * inv_denom) * v;
  }
  out[b * H_DIM + tid] = cacc;
}

// ---------------------------------------------------------------------------
// Launch
// ---------------------------------------------------------------------------
extern "C" void kernel_launch(void* const* d_in, const int* in_sizes, int n_in,
                              void* d_out, int out_size, void* d_ws, size_t ws_size,
                              hipStream_t stream) {
  const float* s_before = (const float*)d_in[0];   // [1,B,H]
  const float* h_sliced = (const float*)d_in[1];   // [N,B,H]
  const float* h        = (const float*)d_in[2];   // [T,B,H]
  const float* W        = (const float*)d_in[3];   // [H,H]
  const float* bias     = (const float*)d_in[4];   // [H]
  float* out = (float*)d_out;                      // [B,H]

  float* ws      = (float*)d_ws;
  float* partial = ws;                                   // 32 * 16384 floats
  float* hsum    = partial + NCHUNK * (B_DIM * H_DIM);   // 16384 floats
  float* sW      = hsum + (B_DIM * H_DIM);               // 16384 floats

  // Phase 1: stream all of h (512 MB) -> hsum  (dominant, HBM-bound ~23 us)
  dim3 g1(PLANE_F4 / 256, NCHUNK);
  hsum_partial_kernel<<<g1, 256, 0, stream>>>((const float4*)h, (float4*)partial);
  hsum_reduce_kernel<<<PLANE_F4 / 256, 256, 0, stream>>>(
      (const float4*)partial, (float4*)hsum);

  // Phase 2: the one true GEMM, on the f32 WMMA path
  sw_wmma_kernel<<<8, 256, 0, stream>>>(s_before, W, bias, sW);

  // Phase 3: fused denom + scores + context (8 MB of h_sliced, read once)
  attn_fused_kernel<<<B_DIM, 256, 0, stream>>>(sW, hsum, h_sliced, out);
}